// GraphEncoder_40011915329829
// MI455X (gfx1250) — compile-verified
//
#include <hip/hip_runtime.h>
#include <hip/hip_bf16.h>

#define N_NODES   100000
#define N_EDGES   3200000
#define D_FEAT    64
#define HIDDEN    16
#define N_GRAPHS  512
#define BN_EPS    1e-5f

typedef __attribute__((ext_vector_type(2))) float v2f;
typedef __attribute__((ext_vector_type(8))) float v8f;

// ---------------------------------------------------------------------------
// Utility kernels
// ---------------------------------------------------------------------------
__global__ void copy_f4(const float4* __restrict__ s, float4* __restrict__ d, int n4) {
  int i = blockIdx.x * blockDim.x + threadIdx.x;
  int stride = gridDim.x * blockDim.x;
  for (; i < n4; i += stride) d[i] = s[i];
}

__global__ void zero_f(float* __restrict__ p, int n) {
  int i = blockIdx.x * blockDim.x + threadIdx.x;
  if (i < n) p[i] = 0.f;
}

// ---------------------------------------------------------------------------
// Edge scatter: agg[dst] += h[src].  One thread per (edge, quad-of-4-features);
// float4 gather + 4 global f32 atomics, grid-stride.
// ---------------------------------------------------------------------------
__global__ void edge_scatter(const int* __restrict__ src, const int* __restrict__ dst,
                             const float* __restrict__ h, float* __restrict__ agg,
                             int nEdges, int D, int lq, int qmask) {
  long long total  = (long long)nEdges << lq;
  long long i      = (long long)blockIdx.x * blockDim.x + threadIdx.x;
  long long stride = (long long)gridDim.x * blockDim.x;
  for (; i < total; i += stride) {
    int e = (int)(i >> lq);
    int q = ((int)i & qmask) << 2;
    int es = src[e], ed = dst[e];
    const float4 v = *(const float4*)(h + (size_t)es * D + q);
    float* ap = agg + (size_t)ed * D + q;
    atomicAdd(ap + 0, v.x);
    atomicAdd(ap + 1, v.y);
    atomicAdd(ap + 2, v.z);
    atomicAdd(ap + 3, v.w);
  }
}

// ---------------------------------------------------------------------------
// GEMM1: z = A[N,K] @ W[K,16] + b  via V_WMMA_F32_16X16X4_F32, plus per-block
// BN partial sums (sum, sum-of-squares) reduced through LDS into global sums.
// One wave owns a 16-row tile.  Grid chosen so every wave is fully active
// (EXEC must be all-ones for WMMA).
// A-operand ISA layout (32-bit A 16x4): lanes 0-15 hold K=0(v0),K=1(v1);
// lanes 16-31 hold K=2(v0),K=3(v1).  B/C/D use the row-striped layout.
// ---------------------------------------------------------------------------
__global__ void gin_gemm1_bnstat(const float* __restrict__ A, int K,
                                 const float* __restrict__ W,
                                 const float* __restrict__ bias,
                                 float* __restrict__ z,
                                 float* __restrict__ sums) {
  const int lane  = threadIdx.x & 31;
  const int wave  = threadIdx.x >> 5;
  const int tile  = blockIdx.x * (blockDim.x >> 5) + wave;
  const int row   = lane & 15;
  const int col   = lane & 15;
  const int khalf = (lane >> 4) << 1;   // 0 or 2
  const int rbase = (lane >> 4) << 3;   // 0 or 8 (C/D row group)

  const float* Arow = A + (size_t)(tile * 16 + row) * K;

  v8f c = {0.f, 0.f, 0.f, 0.f, 0.f, 0.f, 0.f, 0.f};
  for (int kb = 0; kb < K; kb += 4) {
    const int k0 = kb + khalf, k1 = kb + khalf + 1;
    v2f a, b;
    a.x = Arow[k0];
    a.y = Arow[k1];
    b.x = W[k0 * HIDDEN + col];
    b.y = W[k1 * HIDDEN + col];
    c = __builtin_amdgcn_wmma_f32_16x16x4_f32(false, a, false, b, (short)0, c,
                                              false, false);
  }

  const float bv = bias[col];
  float s1 = 0.f, s2 = 0.f;
#pragma unroll
  for (int i = 0; i < 8; ++i) {
    float v = c[i] + bv;
    z[(size_t)(tile * 16 + rbase + i) * HIDDEN + col] = v;
    s1 += v;
    s2 += v * v;
  }

  __shared__ float ls[2 * HIDDEN];
  if (threadIdx.x < 2 * HIDDEN) ls[threadIdx.x] = 0.f;
  __syncthreads();
  atomicAdd(&ls[col], s1);
  atomicAdd(&ls[HIDDEN + col], s2);
  __syncthreads();
  if (threadIdx.x < 2 * HIDDEN) atomicAdd(&sums[threadIdx.x], ls[threadIdx.x]);
}

// ---------------------------------------------------------------------------
// BN finalize: scale = g * rsqrt(var+eps), shift = be - mu*scale
// ---------------------------------------------------------------------------
__global__ void bn_finalize(const float* __restrict__ sums,
                            const float* __restrict__ g,
                            const float* __restrict__ be,
                            float* __restrict__ bn) {
  int cidx = threadIdx.x;
  if (cidx < HIDDEN) {
    const float invN = 1.f / (float)N_NODES;
    float mu  = sums[cidx] * invN;
    float var = sums[HIDDEN + cidx] * invN - mu * mu;
    float sc  = g[cidx] * rsqrtf(var + BN_EPS);
    bn[cidx] = sc;
    bn[HIDDEN + cidx] = be[cidx] - mu * sc;
  }
}

// ---------------------------------------------------------------------------
// GEMM2: h = [relu_out]( relu(z*scale+shift) @ W2 + b2 ); fused BN+ReLU on the
// A operand; writes h and (optionally) the next layer's agg (= self term).
// ---------------------------------------------------------------------------
__global__ void gin_gemm2(const float* __restrict__ z,
                          const float* __restrict__ bn,
                          const float* __restrict__ W2,
                          const float* __restrict__ b2,
                          float* __restrict__ hout,
                          float* __restrict__ aggout,   // may be null
                          int relu_out) {
  const int lane  = threadIdx.x & 31;
  const int wave  = threadIdx.x >> 5;
  const int tile  = blockIdx.x * (blockDim.x >> 5) + wave;
  const int row   = lane & 15;
  const int col   = lane & 15;
  const int khalf = (lane >> 4) << 1;
  const int rbase = (lane >> 4) << 3;

  const float* Zrow = z + (size_t)(tile * 16 + row) * HIDDEN;

  v8f c = {0.f, 0.f, 0.f, 0.f, 0.f, 0.f, 0.f, 0.f};
  for (int kb = 0; kb < HIDDEN; kb += 4) {
    const int k0 = kb + khalf, k1 = kb + khalf + 1;
    v2f a, b;
    a.x = fmaxf(Zrow[k0] * bn[k0] + bn[HIDDEN + k0], 0.f);
    a.y = fmaxf(Zrow[k1] * bn[k1] + bn[HIDDEN + k1], 0.f);
    b.x = W2[k0 * HIDDEN + col];
    b.y = W2[k1 * HIDDEN + col];
    c = __builtin_amdgcn_wmma_f32_16x16x4_f32(false, a, false, b, (short)0, c,
                                              false, false);
  }

  const float bv = b2[col];
#pragma unroll
  for (int i = 0; i < 8; ++i) {
    float v = c[i] + bv;
    if (relu_out) v = fmaxf(v, 0.f);
    size_t idx = (size_t)(tile * 16 + rbase + i) * HIDDEN + col;
    hout[idx] = v;
    if (aggout) aggout[idx] = v;
  }
}

// ---------------------------------------------------------------------------
// Pooling
// ---------------------------------------------------------------------------
__global__ void pool_accum(const float* __restrict__ h, const int* __restrict__ batch,
                           float* __restrict__ pooled, float* __restrict__ counts) {
  int i = blockIdx.x * blockDim.x + threadIdx.x;
  if (i >= N_NODES * HIDDEN) return;
  int n = i >> 4, cix = i & 15;
  int g = batch[n];
  atomicAdd(&pooled[g * HIDDEN + cix], h[i]);
  if (cix == 0) atomicAdd(&counts[g], 1.f);
}

__global__ void pool_norm(float* __restrict__ pooled, const float* __restrict__ counts) {
  int i = blockIdx.x * blockDim.x + threadIdx.x;
  if (i < N_GRAPHS * HIDDEN) pooled[i] /= fmaxf(counts[i >> 4], 1.f);
}

__global__ void out_gather(const float* __restrict__ pooled, const int* __restrict__ batch,
                           float* __restrict__ out) {
  int i = blockIdx.x * blockDim.x + threadIdx.x;
  if (i >= N_NODES * HIDDEN) return;
  int n = i >> 4, cix = i & 15;
  out[i] = pooled[(size_t)batch[n] * HIDDEN + cix];
}

// ---------------------------------------------------------------------------
// Launch
// ---------------------------------------------------------------------------
extern "C" void kernel_launch(void* const* d_in, const int* in_sizes, int n_in,
                              void* d_out, int out_size, void* d_ws, size_t ws_size,
                              hipStream_t stream) {
  (void)in_sizes; (void)n_in; (void)out_size; (void)ws_size;

  const float* x     = (const float*)d_in[0];
  const int*   ei    = (const int*)d_in[1];
  const int*   batch = (const int*)d_in[2];
  const float* W1_0  = (const float*)d_in[3];
  const float* b1_0  = (const float*)d_in[4];
  const float* g0    = (const float*)d_in[5];
  const float* be0   = (const float*)d_in[6];
  const float* W2_0  = (const float*)d_in[7];
  const float* b2_0  = (const float*)d_in[8];
  const float* Ws1   = (const float*)d_in[9];
  const float* bs1   = (const float*)d_in[10];
  const float* gs    = (const float*)d_in[11];
  const float* bes   = (const float*)d_in[12];
  const float* Ws2   = (const float*)d_in[13];
  const float* bs2   = (const float*)d_in[14];

  const int* src = ei;
  const int* dst = ei + N_EDGES;

  // Workspace carving (all regions fully initialized by kernels every call)
  char*  ws  = (char*)d_ws;
  size_t off = 0;
  auto carve = [&](size_t bytes) -> void* {
    void* p = ws + off;
    off += (bytes + 255) & ~(size_t)255;
    return p;
  };
  float* agg64 = (float*)carve((size_t)N_NODES * D_FEAT * sizeof(float));
  float* h     = (float*)carve((size_t)N_NODES * HIDDEN * sizeof(float));
  float* agg16 = (float*)carve((size_t)N_NODES * HIDDEN * sizeof(float));
  float* zbuf  = (float*)carve((size_t)N_NODES * HIDDEN * sizeof(float));
  float* sums  = (float*)carve(2 * HIDDEN * sizeof(float));
  float* bnp   = (float*)carve(2 * HIDDEN * sizeof(float));
  float* pooled= (float*)carve((size_t)N_GRAPHS * HIDDEN * sizeof(float));
  float* counts= (float*)carve((size_t)N_GRAPHS * sizeof(float));

  // GEMM grid: 6250 tiles of 16 rows, 2 waves (64 threads) per block -> exact.
  const int GB = (N_NODES / 16) / 2;  // 3125
  const int GT = 64;
  const int EB = 4096, ET = 256;      // edge-scatter grid-stride

  // ---- Layer 0 (D_FEAT -> HIDDEN) ----
  copy_f4<<<2048, 256, 0, stream>>>((const float4*)x, (float4*)agg64,
                                    N_NODES * D_FEAT / 4);
  edge_scatter<<<EB, ET, 0, stream>>>(src, dst, x, agg64, N_EDGES, D_FEAT, 4, 15);
  zero_f<<<1, 32, 0, stream>>>(sums, 2 * HIDDEN);
  gin_gemm1_bnstat<<<GB, GT, 0, stream>>>(agg64, D_FEAT, W1_0, b1_0, zbuf, sums);
  bn_finalize<<<1, 16, 0, stream>>>(sums, g0, be0, bnp);
  gin_gemm2<<<GB, GT, 0, stream>>>(zbuf, bnp, W2_0, b2_0, h, agg16, 1);

  // ---- Layers 1..9 (HIDDEN -> HIDDEN); layer 9 has no outer ReLU ----
  for (int l = 0; l < 9; ++l) {
    const float* W1 = Ws1 + (size_t)l * HIDDEN * HIDDEN;
    const float* b1 = bs1 + (size_t)l * HIDDEN;
    const float* g  = gs  + (size_t)l * HIDDEN;
    const float* be = bes + (size_t)l * HIDDEN;
    const float* W2 = Ws2 + (size_t)l * HIDDEN * HIDDEN;
    const float* b2 = bs2 + (size_t)l * HIDDEN;
    const bool last = (l == 8);

    edge_scatter<<<EB, ET, 0, stream>>>(src, dst, h, agg16, N_EDGES, HIDDEN, 2, 3);
    zero_f<<<1, 32, 0, stream>>>(sums, 2 * HIDDEN);
    gin_gemm1_bnstat<<<GB, GT, 0, stream>>>(agg16, HIDDEN, W1, b1, zbuf, sums);
    bn_finalize<<<1, 16, 0, stream>>>(sums, g, be, bnp);
    gin_gemm2<<<GB, GT, 0, stream>>>(zbuf, bnp, W2, b2, h,
                                     last ? nullptr : agg16, last ? 0 : 1);
  }

  // ---- Pool + broadcast back to nodes ----
  zero_f<<<(N_GRAPHS * HIDDEN + 255) / 256, 256, 0, stream>>>(pooled, N_GRAPHS * HIDDEN);
  zero_f<<<(N_GRAPHS + 255) / 256, 256, 0, stream>>>(counts, N_GRAPHS);
  pool_accum<<<(N_NODES * HIDDEN + 255) / 256, 256, 0, stream>>>(h, batch, pooled, counts);
  pool_norm<<<(N_GRAPHS * HIDDEN + 255) / 256, 256, 0, stream>>>(pooled, counts);
  out_gather<<<(N_NODES * HIDDEN + 255) / 256, 256, 0, stream>>>(pooled, batch,
                                                                 (float*)d_out);
}